// TensorProduct_86397562126686
// MI455X (gfx1250) — compile-verified
//
#include <hip/hip_runtime.h>
#include <cstdint>

typedef float f4 __attribute__((ext_vector_type(4)));

#define C 128
#define ROWS_PER_BLOCK 8

__device__ __forceinline__ void async_b128_to_lds(uint32_t lds_off, const void* gaddr) {
    // GV addressing: vdst = LDS byte address, vaddr = 64-bit global address.
    asm volatile("global_load_async_to_lds_b128 %0, %1, off"
                 :: "v"(lds_off), "v"(gaddr)
                 : "memory");
}

__device__ __forceinline__ void wait_asynccnt0() {
    asm volatile("s_wait_asynccnt 0" ::: "memory");
}

__global__ __launch_bounds__(256) void TensorProduct_86397562126686_kernel(
    const float* __restrict__ x1,      // N x 512
    const float* __restrict__ x2,      // N x 4
    const float* __restrict__ wgt,     // N x 640
    float* __restrict__ out,           // N x 896
    int N) {
    constexpr float PW2 = 0.70710678118654752f;  // 1/sqrt(2)
    constexpr float IS3 = 0.57735026918962576f;  // 1/sqrt(3)

    __shared__ float s_x2[ROWS_PER_BLOCK][4];

    const int tid  = threadIdx.x;
    const int row0 = blockIdx.x * ROWS_PER_BLOCK;

    // --- Async-stage this block's 8 rows of x2 (16 B each) into LDS ---
    if (tid < ROWS_PER_BLOCK) {
        const int r = row0 + tid;
        if (r < N) {
            const uint32_t lds_off = (uint32_t)(uintptr_t)(&s_x2[tid][0]);
            async_b128_to_lds(lds_off, x2 + (size_t)r * 4);
        }
    }
    wait_asynccnt0();      // issuing wave drains ASYNCcnt; others: no-op (cnt==0)
    __syncthreads();       // LDS now visible to all 8 waves

    const int wave = tid >> 5;   // one wave per row
    const int lane = tid & 31;   // 4 channels per lane
    const int n = row0 + wave;
    if (n >= N) return;

    const float s2  = s_x2[wave][0];   // LDS broadcast reads (same addr per wave)
    const float v2x = s_x2[wave][1];
    const float v2y = s_x2[wave][2];
    const float v2z = s_x2[wave][3];

    // ---- 128-bit non-temporal streaming loads (all 16B-aligned) ----
    const f4* x1s = (const f4*)(x1 + (size_t)n * (4 * C));          // s1 quads
    const f4  s1  = __builtin_nontemporal_load(x1s + lane);

    const f4* v1p = (const f4*)(x1 + (size_t)n * (4 * C) + C + 12 * lane);
    union { f4 q[3]; float f[12]; } v1u;
    v1u.q[0] = __builtin_nontemporal_load(v1p + 0);
    v1u.q[1] = __builtin_nontemporal_load(v1p + 1);
    v1u.q[2] = __builtin_nontemporal_load(v1p + 2);

    const f4* wp = (const f4*)(wgt + (size_t)n * (5 * C));
    const f4 w0 = __builtin_nontemporal_load(wp +   0 + lane);
    const f4 w1 = __builtin_nontemporal_load(wp +  32 + lane);
    const f4 w2 = __builtin_nontemporal_load(wp +  64 + lane);
    const f4 w3 = __builtin_nontemporal_load(wp +  96 + lane);
    const f4 w4 = __builtin_nontemporal_load(wp + 128 + lane);

    // ---- Compute 4 channels ----
    f4 outs;
    union { f4 q[3]; float f[12]; } vo, ve;
#pragma unroll
    for (int j = 0; j < 4; ++j) {
        const float ax = v1u.f[3 * j + 0];
        const float ay = v1u.f[3 * j + 1];
        const float az = v1u.f[3 * j + 2];

        const float dot = ax * v2x + ay * v2y + az * v2z;
        outs[j] = PW2 * (w0[j] * s1[j] * s2 + w1[j] * IS3 * dot);

        vo.f[3 * j + 0] = PW2 * (w2[j] * s1[j] * v2x + w3[j] * ax * s2);
        vo.f[3 * j + 1] = PW2 * (w2[j] * s1[j] * v2y + w3[j] * ay * s2);
        vo.f[3 * j + 2] = PW2 * (w2[j] * s1[j] * v2z + w3[j] * az * s2);

        ve.f[3 * j + 0] = PW2 * w4[j] * (ay * v2z - az * v2y);
        ve.f[3 * j + 1] = PW2 * w4[j] * (az * v2x - ax * v2z);
        ve.f[3 * j + 2] = PW2 * w4[j] * (ax * v2y - ay * v2x);
    }

    // ---- 128-bit non-temporal streaming stores (all 16B-aligned) ----
    float* outr = out + (size_t)n * (7 * C);
    __builtin_nontemporal_store(outs, (f4*)outr + lane);

    f4* vop = (f4*)(outr + C + 12 * lane);
    __builtin_nontemporal_store(vo.q[0], vop + 0);
    __builtin_nontemporal_store(vo.q[1], vop + 1);
    __builtin_nontemporal_store(vo.q[2], vop + 2);

    f4* vep = (f4*)(outr + 4 * C + 12 * lane);
    __builtin_nontemporal_store(ve.q[0], vep + 0);
    __builtin_nontemporal_store(ve.q[1], vep + 1);
    __builtin_nontemporal_store(ve.q[2], vep + 2);
}

extern "C" void kernel_launch(void* const* d_in, const int* in_sizes, int n_in,
                              void* d_out, int out_size, void* d_ws, size_t ws_size,
                              hipStream_t stream) {
    const float* x1  = (const float*)d_in[0];
    const float* x2  = (const float*)d_in[1];
    const float* wgt = (const float*)d_in[2];
    float* out = (float*)d_out;

    const int N = in_sizes[0] / (4 * C);   // x1 is N x 512
    const int blocks = (N + ROWS_PER_BLOCK - 1) / ROWS_PER_BLOCK;
    TensorProduct_86397562126686_kernel<<<blocks, 256, 0, stream>>>(x1, x2, wgt, out, N);
}